// GraphAttentionLayer_13683765805323
// MI455X (gfx1250) — compile-verified
//
#include <hip/hip_runtime.h>
#include <hip/hip_bf16.h>
#include <math.h>

#define BB 8
#define NN 2048
#define IN_DIM 128
#define OUT_DIM 64
#define NEG_SLOPE 0.2f

typedef __attribute__((ext_vector_type(2))) float v2f;
typedef __attribute__((ext_vector_type(8))) float v8f;

// ---------------------------------------------------------------------------
// Kernel 1: Wh = X @ W  (fp32 WMMA 16x16x4). One wave = 16 rows x 64 cols.
// W is staged once in LDS in a row-pair-interleaved layout so each B fragment
// is a single aligned ds_load_b64 into an even VGPR pair (no repack movs).
//   layout: wlds[(k/2)*128 + 2*c + (k&1)]  for element W[k][c]
// ---------------------------------------------------------------------------
__global__ __launch_bounds__(128) void gat_wh(const float* __restrict__ X,
                                              const float* __restrict__ W,
                                              float* __restrict__ Wh) {
  __shared__ float wlds[IN_DIM * OUT_DIM];  // 32 KB, shared by 4 waves
  const int t = threadIdx.x;

  // cooperative interleaved stage: 64 row-pairs x 32 col-pairs = 2048 units
  for (int u = t; u < (IN_DIM / 2) * (OUT_DIM / 2); u += 128) {
    const int p = u >> 5, cg = u & 31;
    const float* g = W + p * 2 * OUT_DIM + cg * 2;
    const float2 r0 = *(const float2*)g;
    const float2 r1 = *(const float2*)(g + OUT_DIM);
    *(float4*)&wlds[p * 128 + cg * 4] = make_float4(r0.x, r1.x, r0.y, r1.y);
  }
  __syncthreads();

  const int wave = t >> 5, lane = t & 31;
  const int half = lane >> 4, l16 = lane & 15;
  const long row0 = (long)blockIdx.x * 64 + wave * 16;

  v8f acc[4] = {};
  const float* xrow = X + (row0 + l16) * IN_DIM;

  for (int kk = 0; kk < IN_DIM / 4; ++kk) {
    const int k0 = kk * 4 + half * 2;   // A layout: lanes 0-15 K={4kk,4kk+1}, 16-31 {+2,+3}
    const int prow = kk * 2 + half;     // interleaved row-pair index (k0/2)
    v2f afrag;
    afrag.x = xrow[k0];
    afrag.y = xrow[k0 + 1];
#pragma unroll
    for (int nt = 0; nt < 4; ++nt) {
      const v2f bfrag = *(const v2f*)&wlds[prow * 128 + 2 * (nt * 16 + l16)];
      acc[nt] = __builtin_amdgcn_wmma_f32_16x16x4_f32(
          false, afrag, false, bfrag, (short)0, acc[nt], false, false);
    }
  }
  // C layout: VGPR r -> M = r + 8*half, N = l16
#pragma unroll
  for (int nt = 0; nt < 4; ++nt)
#pragma unroll
    for (int r = 0; r < 8; ++r)
      Wh[(row0 + r + half * 8) * OUT_DIM + nt * 16 + l16] = acc[nt][r];
}

// ---------------------------------------------------------------------------
// Kernel 2: src = Wh @ a[:64], dst = Wh @ a[64:]. One wave per row.
// ---------------------------------------------------------------------------
__global__ __launch_bounds__(256) void gat_srcdst(const float* __restrict__ Wh,
                                                  const float* __restrict__ avec,
                                                  float* __restrict__ srcb,
                                                  float* __restrict__ dstb) {
  const int wave = threadIdx.x >> 5, lane = threadIdx.x & 31;
  const long row = (long)blockIdx.x * 8 + wave;
  const float* wr = Wh + row * OUT_DIM;
  const float w0 = wr[lane], w1 = wr[lane + 32];
  float ps = w0 * avec[lane] + w1 * avec[lane + 32];
  float pd = w0 * avec[64 + lane] + w1 * avec[96 + lane];
  for (int off = 16; off >= 1; off >>= 1) {
    ps += __shfl_xor(ps, off, 32);
    pd += __shfl_xor(pd, off, 32);
  }
  if (lane == 0) { srcb[row] = ps; dstb[row] = pd; }
}

// ---------------------------------------------------------------------------
// Kernel 3: per-row masked-softmax stats (m, l) via online update + shuffle.
// ---------------------------------------------------------------------------
__global__ __launch_bounds__(256) void gat_stats(const float* __restrict__ A,
                                                 const float* __restrict__ srcb,
                                                 const float* __restrict__ dstb,
                                                 float* __restrict__ mrow,
                                                 float* __restrict__ lrow) {
  const int wave = threadIdx.x >> 5, lane = threadIdx.x & 31;
  const int row = blockIdx.x * 8 + wave;
  const int b = row >> 11, i = row & (NN - 1);
  const float sv = srcb[row];
  const float* Ar = A + (long)i * NN;
  const float* dr = dstb + (long)b * NN;

  float mm = -INFINITY, ss = 0.f;
  for (int j = lane; j < NN; j += 32) {
    const float av = Ar[j];
    if (av != 0.f) {
      float e = sv + dr[j];
      e = (e >= 0.f) ? e : NEG_SLOPE * e;
      if (e > mm) { ss = ss * __expf(mm - e) + 1.f; mm = e; }
      else        { ss += __expf(e - mm); }
    }
  }
  for (int off = 16; off >= 1; off >>= 1) {
    const float om = __shfl_xor(mm, off, 32);
    const float os = __shfl_xor(ss, off, 32);
    const float nm = (mm > om) ? mm : om;
    const float s1 = (mm == nm) ? ss : ss * __expf(mm - nm);  // -inf safe
    const float s2 = (om == nm) ? os : os * __expf(om - nm);
    ss = s1 + s2;
    mm = nm;
  }
  if (lane == 0) { mrow[row] = mm; lrow[row] = ss; }
}

// ---------------------------------------------------------------------------
// Kernel 4: out = ELU( (softmax(e)*A) @ Wh ), flash-style fused. One wave per
// 16-row i-tile; 4 waves share a double-buffered interleaved Wh j-tile in LDS.
// ---------------------------------------------------------------------------
__device__ __forceinline__ float edge_p(float av, float sv, float dj,
                                        float mi, float invl) {
  if (av == 0.f) return 0.f;
  float e = sv + dj;
  e = (e >= 0.f) ? e : NEG_SLOPE * e;
  return __expf(e - mi) * invl * av;
}

__global__ __launch_bounds__(128) void gat_attn(const float* __restrict__ A,
                                                const float* __restrict__ Wh,
                                                const float* __restrict__ srcb,
                                                const float* __restrict__ dstb,
                                                const float* __restrict__ mrow,
                                                const float* __restrict__ lrow,
                                                float* __restrict__ out) {
  // double-buffered 16x64 tile, row-pair-interleaved: [p*128 + 2c + (row&1)]
  __shared__ float whs[2][16 * OUT_DIM];  // 2 x 4 KB
  const int t = threadIdx.x;
  const int wave = t >> 5, lane = t & 31;
  const int half = lane >> 4, l16 = lane & 15;
  const int b = blockIdx.x >> 5;                 // 32 blocks per batch
  const int i0 = (blockIdx.x & 31) * 64 + wave * 16;
  const int il = i0 + l16;
  const long rrow = (long)b * NN + il;

  const float sv = srcb[rrow];
  const float mi = mrow[rrow];
  const float li = lrow[rrow];
  const float invl = (li > 0.f) ? 1.f / li : 0.f;

  const float* Ar = A + (long)il * NN;
  const float* dr = dstb + (long)b * NN;
  const float* whb = Wh + (long)b * NN * OUT_DIM;

  v8f acc[4] = {};

  // stage(jt -> buf): 8 row-pairs x 32 col-pairs = 256 units, 2 per thread
  auto stage = [&](int buf, int jt) {
#pragma unroll
    for (int u0 = 0; u0 < 2; ++u0) {
      const int u = t + u0 * 128;
      const int p = u >> 5, cg = u & 31;
      const float* g = whb + (long)jt * 16 * OUT_DIM + p * 2 * OUT_DIM + cg * 2;
      const float2 r0 = *(const float2*)g;
      const float2 r1 = *(const float2*)(g + OUT_DIM);
      *(float4*)&whs[buf][p * 128 + cg * 4] = make_float4(r0.x, r1.x, r0.y, r1.y);
    }
  };

  stage(0, 0);
  __syncthreads();

  for (int jt = 0; jt < NN / 16; ++jt) {
    const int cur = jt & 1;
    if (jt + 1 < NN / 16) {
      stage(cur ^ 1, jt + 1);                       // overlaps with WMMA below
      __builtin_prefetch(Ar + (jt + 1) * 16, 0, 1); // next A-row segment
    }

#pragma unroll
    for (int kk = 0; kk < 4; ++kk) {
      const int jr = kk * 4 + half * 2;   // even
      const int j = jt * 16 + jr;
      const int prow = kk * 2 + half;     // interleaved row-pair index
      v2f p;  // A-operand fragment of P (rows = i, K = j)
      p.x = edge_p(Ar[j],     sv, dr[j],     mi, invl);
      p.y = edge_p(Ar[j + 1], sv, dr[j + 1], mi, invl);
#pragma unroll
      for (int nt = 0; nt < 4; ++nt) {
        const v2f bf = *(const v2f*)&whs[cur][prow * 128 + 2 * (nt * 16 + l16)];
        acc[nt] = __builtin_amdgcn_wmma_f32_16x16x4_f32(
            false, p, false, bf, (short)0, acc[nt], false, false);
      }
    }
    __syncthreads();
  }

  // fused ELU + store
  float* orow = out + ((long)b * NN + i0 + half * 8) * OUT_DIM;
#pragma unroll
  for (int nt = 0; nt < 4; ++nt)
#pragma unroll
    for (int r = 0; r < 8; ++r) {
      const float x = acc[nt][r];
      const float y = (x > 0.f) ? x : (__expf(x) - 1.f);
      orow[(long)r * OUT_DIM + nt * 16 + l16] = y;
    }
}

// ---------------------------------------------------------------------------
extern "C" void kernel_launch(void* const* d_in, const int* in_sizes, int n_in,
                              void* d_out, int out_size, void* d_ws, size_t ws_size,
                              hipStream_t stream) {
  const float* X = (const float*)d_in[0];   // (8,2048,128)
  const float* A = (const float*)d_in[1];   // (2048,2048)
  const float* W = (const float*)d_in[2];   // (128,64)
  const float* a = (const float*)d_in[3];   // (128,)
  float* out = (float*)d_out;               // (8,2048,64)

  // workspace layout (floats): Wh | src | dst | m | l  (~4.3 MB total)
  float* Wh   = (float*)d_ws;
  float* srcb = Wh + (size_t)BB * NN * OUT_DIM;
  float* dstb = srcb + (size_t)BB * NN;
  float* mrow = dstb + (size_t)BB * NN;
  float* lrow = mrow + (size_t)BB * NN;

  // 1) Wh = X @ W
  gat_wh<<<(BB * NN) / 64, 128, 0, stream>>>(X, W, Wh);
  // 2) src/dst projections
  gat_srcdst<<<(BB * NN) / 8, 256, 0, stream>>>(Wh, a, srcb, dstb);
  // 3) softmax stats
  gat_stats<<<(BB * NN) / 8, 256, 0, stream>>>(A, srcb, dstb, mrow, lrow);
  // 4) fused attention aggregation + ELU
  gat_attn<<<BB * (NN / 64), 128, 0, stream>>>(A, Wh, srcb, dstb, mrow, lrow, out);
}